// GATNet_74354473828736
// MI455X (gfx1250) — compile-verified
//
#include <hip/hip_runtime.h>
#include <hip/hip_bf16.h>
#include <math.h>

// ---------------------------------------------------------------------------
// 2-layer GAT (PyG GATConv semantics) for MI455X / gfx1250.
//   L1: h1 = x@W1 [50000,128]; 4-head attention, segment softmax over dst,
//       scatter-aggregate, +b1, ELU.
//   L2: h2 = hE@W2 [50000,16]; 1-head attention, segment softmax, aggregate,
//       +b2, log_softmax.
// GEMMs: native fp32 WMMA (v_wmma_f32_16x16x4_f32), weights staged in LDS
// via GLOBAL_LOAD_ASYNC_TO_LDS_B128 (fallback: plain float4 copy), one wave
// per 16-row strip reusing its A-fragment across all N-tiles.
// Edge passes are L2-resident (192 MB L2) and use f32 atomics.
// ---------------------------------------------------------------------------

#define N_NODES   50000
#define N_EDGES   1600000
#define E_TOT     (N_EDGES + N_NODES)   // edges + self loops
#define IN_C      256
#define HID       32
#define HEADS     4
#define F1        (HEADS * HID)         // 128
#define NCLS      16
#define NEG_SLOPE 0.2f

typedef float v2f __attribute__((ext_vector_type(2)));
typedef float v8f __attribute__((ext_vector_type(8)));
typedef int   v4i __attribute__((ext_vector_type(4)));
typedef __attribute__((address_space(1))) v4i gv4i;   // global int4
typedef __attribute__((address_space(3))) v4i lv4i;   // LDS int4 (32-bit ptr)

#if __has_builtin(__builtin_amdgcn_global_load_async_to_lds_b128)
#define HAVE_ASYNC_TO_LDS 1
#else
#define HAVE_ASYNC_TO_LDS 0
#endif

// ---- cooperative global->LDS stage (async path when available) ------------
__device__ __forceinline__ void fill_lds_f4(float* __restrict__ dst,
                                            const float* __restrict__ src,
                                            int nfloats, int tid, int nthr) {
#if HAVE_ASYNC_TO_LDS
  for (int i = tid * 4; i < nfloats; i += nthr * 4) {
    gv4i* g = (gv4i*)(unsigned long long)(size_t)(src + i);
    lv4i* l = (lv4i*)(unsigned int)(size_t)(dst + i);
    __builtin_amdgcn_global_load_async_to_lds_b128(g, l, 0, 0);
  }
#if __has_builtin(__builtin_amdgcn_s_wait_asynccnt)
  __builtin_amdgcn_s_wait_asynccnt(0);
#else
  asm volatile("s_wait_asynccnt 0x0" ::: "memory");
#endif
#else
  for (int i = tid * 4; i < nfloats; i += nthr * 4)
    *reinterpret_cast<float4*>(dst + i) = *reinterpret_cast<const float4*>(src + i);
#endif
}

// ---- order-preserving float<->uint map (for atomicMax on floats) ----------
__device__ __forceinline__ unsigned f2ord(float f) {
  unsigned u = __float_as_uint(f);
  return (u & 0x80000000u) ? ~u : (u | 0x80000000u);
}
__device__ __forceinline__ float ord2f(unsigned u) {
  return (u & 0x80000000u) ? __uint_as_float(u & 0x7FFFFFFFu)
                           : __uint_as_float(~u);
}
// note: 0u is a valid "-inf-or-lower" init (all real encodings > 0x007FFFFF)

__device__ __forceinline__ void edge_sd(const int* __restrict__ ei, long long e,
                                        int& s, int& d) {
  if (e < (long long)N_EDGES) { s = ei[e]; d = ei[(long long)N_EDGES + e]; }
  else                        { s = d = (int)(e - (long long)N_EDGES); }
}

__device__ __forceinline__ float leaky(float v) {
  return v > 0.f ? v : NEG_SLOPE * v;
}

// ---------------------------------------------------------------------------
// fp32 WMMA GEMM, LDS-staged B: C[M,NCOL] = A[M,K] * B[K,NCOL].
// One wave computes a full 16 x NCOL strip: the A-fragment (16x4 f32) is
// loaded once per K-step and reused for every 16-col N-tile from LDS.
// A-frag 16x4: lane m=lane&15, holds K = 2*(lane>>4) + {0,1}.
// B-frag 4x16: lane n=lane&15, holds K = 2*(lane>>4) + {0,1}.
// C/D 16x16: VGPR j -> row j (lanes 0-15) / row j+8 (lanes 16-31), col=lane&15.
// ---------------------------------------------------------------------------
template <int K, int NCOL>
__global__ __launch_bounds__(256) void wmma_gemm_lds(const float* __restrict__ A,
                                                     const float* __restrict__ B,
                                                     float* __restrict__ C, int M) {
  constexpr int NT = NCOL / 16;
  __shared__ float sB[K * NCOL];            // GEMM1: 128 KB, GEMM2: 8 KB
  const int tid = threadIdx.x;
  fill_lds_f4(sB, B, K * NCOL, tid, 256);
  __syncthreads();

  const int wave = blockIdx.x * 8 + (tid >> 5);
  const int lane = tid & 31;
  const int r    = lane & 15;
  const int kb   = (lane >> 4) << 1;        // 0 or 2
  if (wave * 16 < M) {
    const float* __restrict__ arow = A + (size_t)(wave * 16 + r) * K;
    v8f acc[NT] = {};
    for (int k0 = 0; k0 < K; k0 += 4) {
      v2f a;
      a.x = arow[k0 + kb];
      a.y = arow[k0 + kb + 1];
      #pragma unroll
      for (int nt = 0; nt < NT; ++nt) {
        v2f b;
        b.x = sB[(k0 + kb) * NCOL + nt * 16 + r];
        b.y = sB[(k0 + kb + 1) * NCOL + nt * 16 + r];
        acc[nt] = __builtin_amdgcn_wmma_f32_16x16x4_f32(false, a, false, b,
                                                        (short)0, acc[nt],
                                                        false, false);
      }
    }
    const int rowbase = wave * 16 + ((lane >> 4) << 3);
    #pragma unroll
    for (int nt = 0; nt < NT; ++nt) {
      const int col = nt * 16 + r;
      #pragma unroll
      for (int j = 0; j < 8; ++j)
        C[(size_t)(rowbase + j) * NCOL + col] = acc[nt][j];
    }
  }
}

// ---- attention logits per node --------------------------------------------
__global__ void alpha1_kernel(const float* __restrict__ h1,
                              const float* __restrict__ as,
                              const float* __restrict__ ad,
                              float* __restrict__ alpha_s,
                              float* __restrict__ alpha_d) {
  int t = blockIdx.x * blockDim.x + threadIdx.x;
  if (t >= N_NODES * HEADS) return;
  int n = t >> 2, h = t & 3;
  const float* row = h1 + (size_t)n * F1 + h * HID;
  float ss = 0.f, sd = 0.f;
  #pragma unroll 8
  for (int c = 0; c < HID; ++c) {
    float v = row[c];
    ss += v * as[h * HID + c];
    sd += v * ad[h * HID + c];
  }
  alpha_s[t] = ss;
  alpha_d[t] = sd;
}

__global__ void alpha2_kernel(const float* __restrict__ h2,
                              const float* __restrict__ as,
                              const float* __restrict__ ad,
                              float* __restrict__ alpha_s,
                              float* __restrict__ alpha_d) {
  int n = blockIdx.x * blockDim.x + threadIdx.x;
  if (n >= N_NODES) return;
  const float* row = h2 + (size_t)n * NCLS;
  float ss = 0.f, sd = 0.f;
  #pragma unroll
  for (int c = 0; c < NCLS; ++c) {
    float v = row[c];
    ss += v * as[c];
    sd += v * ad[c];
  }
  alpha_s[n] = ss;
  alpha_d[n] = sd;
}

// ---- edge pass 1: segment max ---------------------------------------------
template <int H>
__global__ void edge_max_kernel(const int* __restrict__ ei,
                                const float* __restrict__ alpha_s,
                                const float* __restrict__ alpha_d,
                                unsigned* __restrict__ mbuf) {
  long long t = (long long)blockIdx.x * blockDim.x + threadIdx.x;
  if (t >= (long long)E_TOT * H) return;
  long long e = t / H;
  int h = (int)(t - e * H);
  __builtin_prefetch(ei + e + 4096, 0, 0);   // global_prefetch_b8 (speculative)
  int s, d;
  edge_sd(ei, e, s, d);
  float ev = leaky(alpha_s[s * H + h] + alpha_d[d * H + h]);
  atomicMax(&mbuf[d * H + h], f2ord(ev));
}

// ---- edge pass 2: exp + segment sum (stores per-edge numerator) -----------
template <int H>
__global__ void edge_sum_kernel(const int* __restrict__ ei,
                                const float* __restrict__ alpha_s,
                                const float* __restrict__ alpha_d,
                                const unsigned* __restrict__ mbuf,
                                float* __restrict__ pbuf,
                                float* __restrict__ denom) {
  long long t = (long long)blockIdx.x * blockDim.x + threadIdx.x;
  if (t >= (long long)E_TOT * H) return;
  long long e = t / H;
  int h = (int)(t - e * H);
  int s, d;
  edge_sd(ei, e, s, d);
  float ev = leaky(alpha_s[s * H + h] + alpha_d[d * H + h]);
  float m  = ord2f(mbuf[d * H + h]);
  float p  = __expf(ev - m);
  pbuf[t] = p;
  atomicAdd(&denom[d * H + h], p);
}

// ---- edge pass 3 (layer 1): wave per edge, float4 per lane ----------------
__global__ void edge_agg1_kernel(const int* __restrict__ ei,
                                 const float* __restrict__ h1,
                                 const float* __restrict__ pbuf,
                                 const float* __restrict__ denom,
                                 float* __restrict__ out1) {
  long long wave = (long long)blockIdx.x * (blockDim.x >> 5) + (threadIdx.x >> 5);
  if (wave >= (long long)E_TOT) return;
  int lane = threadIdx.x & 31;
  int s, d;
  edge_sd(ei, wave, s, d);
  int h = lane >> 3;  // channels [lane*4, lane*4+3] all belong to head lane/8
  float coef = pbuf[wave * HEADS + h] / (denom[d * HEADS + h] + 1e-16f);
  const float4 hv = *reinterpret_cast<const float4*>(h1 + (size_t)s * F1 + lane * 4);
  float* o = out1 + (size_t)d * F1 + lane * 4;
  atomicAdd(o + 0, hv.x * coef);
  atomicAdd(o + 1, hv.y * coef);
  atomicAdd(o + 2, hv.z * coef);
  atomicAdd(o + 3, hv.w * coef);
}

// ---- edge pass 3 (layer 2): thread per (edge, class) ----------------------
__global__ void edge_agg2_kernel(const int* __restrict__ ei,
                                 const float* __restrict__ h2,
                                 const float* __restrict__ pbuf,
                                 const float* __restrict__ denom,
                                 float* __restrict__ out2) {
  long long t = (long long)blockIdx.x * blockDim.x + threadIdx.x;
  if (t >= (long long)E_TOT * NCLS) return;
  long long e = t >> 4;
  int c = (int)(t & 15);
  int s, d;
  edge_sd(ei, e, s, d);
  float coef = pbuf[e] / (denom[d] + 1e-16f);
  atomicAdd(&out2[(size_t)d * NCLS + c], h2[(size_t)s * NCLS + c] * coef);
}

// ---- +b1 then ELU ----------------------------------------------------------
__global__ void bias_elu_kernel(const float* __restrict__ out1,
                                const float* __restrict__ b1,
                                float* __restrict__ hE) {
  long long t = (long long)blockIdx.x * blockDim.x + threadIdx.x;
  if (t >= (long long)N_NODES * F1) return;
  float v = out1[t] + b1[(int)(t & (F1 - 1))];
  hE[t] = v > 0.f ? v : (__expf(v) - 1.f);
}

// ---- +b2 then row log_softmax ---------------------------------------------
__global__ void final_kernel(const float* __restrict__ out2,
                             const float* __restrict__ b2,
                             float* __restrict__ out) {
  int n = blockIdx.x * blockDim.x + threadIdx.x;
  if (n >= N_NODES) return;
  float v[NCLS];
  float mx = -INFINITY;
  #pragma unroll
  for (int c = 0; c < NCLS; ++c) {
    v[c] = out2[(size_t)n * NCLS + c] + b2[c];
    mx = fmaxf(mx, v[c]);
  }
  float se = 0.f;
  #pragma unroll
  for (int c = 0; c < NCLS; ++c) se += __expf(v[c] - mx);
  float lse = mx + __logf(se);
  #pragma unroll
  for (int c = 0; c < NCLS; ++c) out[(size_t)n * NCLS + c] = v[c] - lse;
}

// ---------------------------------------------------------------------------
extern "C" void kernel_launch(void* const* d_in, const int* in_sizes, int n_in,
                              void* d_out, int out_size, void* d_ws, size_t ws_size,
                              hipStream_t stream) {
  (void)in_sizes; (void)n_in; (void)out_size; (void)ws_size;
  const float* x   = (const float*)d_in[0];
  const int*   ei  = (const int*)d_in[1];   // edge_index [2,E] row-major
  const float* W1  = (const float*)d_in[2];
  const float* as1 = (const float*)d_in[3];
  const float* ad1 = (const float*)d_in[4];
  const float* b1  = (const float*)d_in[5];
  const float* W2  = (const float*)d_in[6];
  const float* as2 = (const float*)d_in[7];
  const float* ad2 = (const float*)d_in[8];
  const float* b2  = (const float*)d_in[9];
  float* out = (float*)d_out;

  // ---- workspace layout (all 16B aligned; ~120 MB total) ----
  float* ws = (float*)d_ws;
  size_t off = 0;
  auto alloc = [&](size_t nf) { float* p = ws + off; off += nf; return p; };
  float* h1      = alloc((size_t)N_NODES * F1);
  float* hE      = alloc((size_t)N_NODES * F1);
  float* h2      = alloc((size_t)N_NODES * NCLS);
  float* alphaS1 = alloc((size_t)N_NODES * HEADS);
  float* alphaD1 = alloc((size_t)N_NODES * HEADS);
  float* alphaS2 = alloc((size_t)N_NODES);
  float* alphaD2 = alloc((size_t)N_NODES);
  float* p1      = alloc((size_t)E_TOT * HEADS);
  float* p2      = alloc((size_t)E_TOT);
  // zero-init region (memset every launch: graph-replay safe)
  float* zbase   = ws + off;
  unsigned* m1   = (unsigned*)alloc((size_t)N_NODES * HEADS);
  float* denom1  = alloc((size_t)N_NODES * HEADS);
  float* out1    = alloc((size_t)N_NODES * F1);
  unsigned* m2   = (unsigned*)alloc((size_t)N_NODES);
  float* denom2  = alloc((size_t)N_NODES);
  float* out2    = alloc((size_t)N_NODES * NCLS);
  size_t zbytes  = (size_t)((ws + off) - zbase) * sizeof(float);
  (void)hipMemsetAsync(zbase, 0, zbytes, stream);

  const int BLK = 256;                        // 8 wave32 per workgroup
  const int waves_per_blk = BLK / 32;
  auto cdiv = [](long long a, long long b) { return (int)((a + b - 1) / b); };

  // ---- layer 1 ----
  {
    int mtiles = (N_NODES + 15) / 16;                        // 3125 strips
    int blocks = cdiv(mtiles, waves_per_blk);
    wmma_gemm_lds<IN_C, F1><<<blocks, BLK, 0, stream>>>(x, W1, h1, N_NODES);
  }
  alpha1_kernel<<<cdiv((long long)N_NODES * HEADS, BLK), BLK, 0, stream>>>(
      h1, as1, ad1, alphaS1, alphaD1);
  edge_max_kernel<HEADS><<<cdiv((long long)E_TOT * HEADS, BLK), BLK, 0, stream>>>(
      ei, alphaS1, alphaD1, m1);
  edge_sum_kernel<HEADS><<<cdiv((long long)E_TOT * HEADS, BLK), BLK, 0, stream>>>(
      ei, alphaS1, alphaD1, m1, p1, denom1);
  edge_agg1_kernel<<<cdiv((long long)E_TOT, waves_per_blk), BLK, 0, stream>>>(
      ei, h1, p1, denom1, out1);
  bias_elu_kernel<<<cdiv((long long)N_NODES * F1, BLK), BLK, 0, stream>>>(
      out1, b1, hE);

  // ---- layer 2 ----
  {
    int mtiles = (N_NODES + 15) / 16;
    int blocks = cdiv(mtiles, waves_per_blk);
    wmma_gemm_lds<F1, NCLS><<<blocks, BLK, 0, stream>>>(hE, W2, h2, N_NODES);
  }
  alpha2_kernel<<<cdiv(N_NODES, BLK), BLK, 0, stream>>>(h2, as2, ad2, alphaS2, alphaD2);
  edge_max_kernel<1><<<cdiv((long long)E_TOT, BLK), BLK, 0, stream>>>(
      ei, alphaS2, alphaD2, m2);
  edge_sum_kernel<1><<<cdiv((long long)E_TOT, BLK), BLK, 0, stream>>>(
      ei, alphaS2, alphaD2, m2, p2, denom2);
  edge_agg2_kernel<<<cdiv((long long)E_TOT * NCLS, BLK), BLK, 0, stream>>>(
      ei, h2, p2, denom2, out2);
  final_kernel<<<cdiv(N_NODES, BLK), BLK, 0, stream>>>(out2, b2, out);
}